// ForwardForwardCoutingLayer_19593640804423
// MI455X (gfx1250) — compile-verified
//
#include <hip/hip_runtime.h>

// ForwardForwardCountingLayer forward: out[b,o] = (op==0 ? min_i : max_i) of
//   vals[b,o,i] = (et==2 ? 1-x[b,i] : x[b,i]) + (et==0 ? (op==0 ? +10 : -10) : 0)
//
// Memory-bound: 256 MB of int32 edge types streamed once per call
// (~11 us floor at 23.3 TB/s). Strategy: 128-bit NT streaming loads of et,
// x row staged to LDS via CDNA5 async global->LDS copy, wave32 ds_swizzle
// reduction, global_prefetch for the next row.

typedef int   v4i __attribute__((ext_vector_type(4)));
typedef float v4f __attribute__((ext_vector_type(4)));

#define FFC_B 64
#define FFC_I 1024
#define FFC_O 1024

#define BLOCK_THREADS 256
#define WAVE_SZ 32
#define OBLK 16                 // outputs per block
#define O_PER_WAVE (OBLK / (BLOCK_THREADS / WAVE_SZ))   // = 2

#define GLOBAL_AS __attribute__((address_space(1)))
#define LDS_AS    __attribute__((address_space(3)))

#if defined(__has_builtin)
#  if __has_builtin(__builtin_amdgcn_global_load_async_to_lds_b128)
#    define HAVE_ASYNC_BUILTIN 1
#  endif
#  if __has_builtin(__builtin_amdgcn_s_wait_asynccnt)
#    define HAVE_ASYNCCNT_BUILTIN 1
#  endif
#endif

// Stage 16 bytes per thread from global to LDS using the CDNA5 async
// global->LDS path (tracked with ASYNCcnt), then wait on own wave's counter.
__device__ __forceinline__ void stage_row_async(const float* __restrict__ grow,
                                                float* __restrict__ ldsrow,
                                                int tid) {
#if defined(HAVE_ASYNC_BUILTIN)
  __builtin_amdgcn_global_load_async_to_lds_b128(
      (GLOBAL_AS v4i*)(grow + tid * 4),
      (LDS_AS v4i*)(ldsrow + tid * 4),
      /*offset=*/0, /*cpol=*/0);
#else
  // saddr form: vdst = LDS byte address, vaddr = per-lane byte offset,
  // saddr = uniform 64-bit base. Low 32 bits of a flat LDS pointer are the
  // LDS byte offset (ISA 10.2 aperture mapping).
  unsigned boff   = (unsigned)tid * 16u;
  unsigned ldsoff = (unsigned)(unsigned long long)(ldsrow) + boff;
  asm volatile("global_load_async_to_lds_b128 %0, %1, %2"
               :: "v"(ldsoff), "v"(boff), "s"(grow) : "memory");
#endif
#if defined(HAVE_ASYNCCNT_BUILTIN)
  __builtin_amdgcn_s_wait_asynccnt(0);
#else
  asm volatile("s_wait_asynccnt 0" ::: "memory");
#endif
}

// wave32 XOR-butterfly via DS_SWIZZLE_B32 group-of-32 mode:
// imm = {1'b0, xor_mask[4:0], or_mask[4:0]=0, and_mask[4:0]=0x1f}
#define SWZ_F32(v, imm) \
  __int_as_float(__builtin_amdgcn_ds_swizzle(__float_as_int(v), (imm)))

__device__ __forceinline__ float wave32_reduce_min(float v) {
  v = fminf(v, SWZ_F32(v, 0x401f));  // xor 16
  v = fminf(v, SWZ_F32(v, 0x201f));  // xor 8
  v = fminf(v, SWZ_F32(v, 0x101f));  // xor 4
  v = fminf(v, SWZ_F32(v, 0x081f));  // xor 2
  v = fminf(v, SWZ_F32(v, 0x041f));  // xor 1
  return v;
}

__device__ __forceinline__ float wave32_reduce_max(float v) {
  v = fmaxf(v, SWZ_F32(v, 0x401f));
  v = fmaxf(v, SWZ_F32(v, 0x201f));
  v = fmaxf(v, SWZ_F32(v, 0x101f));
  v = fmaxf(v, SWZ_F32(v, 0x081f));
  v = fmaxf(v, SWZ_F32(v, 0x041f));
  return v;
}

__global__ __launch_bounds__(BLOCK_THREADS)
void ffc_minmax_kernel(const float* __restrict__ x,
                       const int*   __restrict__ et,
                       const int*   __restrict__ op_idx,
                       float*       __restrict__ out) {
  __shared__ __align__(16) float xs[FFC_I];   // x[b, :]
  __shared__ __align__(16) float oxs[FFC_I];  // 1 - x[b, :]

  const int b    = blockIdx.x;
  const int tid  = threadIdx.x;
  const int lane = tid & (WAVE_SZ - 1);
  const int wave = tid >> 5;

  // ---- stage x[b,:] (4 KB) into LDS with async copy, derive 1-x ----
  const float* xrow = x + (size_t)b * FFC_I;
  stage_row_async(xrow, xs, tid);
  {
    // own wave's async requests are complete; each thread touches only the
    // 16 bytes it loaded itself before the barrier.
    v4f xv = *(const v4f*)(xs + tid * 4);
    *(v4f*)(oxs + tid * 4) = 1.0f - xv;
  }
  __syncthreads();

  const int obase = blockIdx.y * OBLK + wave * O_PER_WAVE;

  for (int j = 0; j < O_PER_WAVE; ++j) {
    const int o = obase + j;
    const int* __restrict__ row = et + ((size_t)b * FFC_O + o) * FFC_I;

    // Prefetch next row (4 KB = 32 lanes x 128B cachelines) while streaming
    // this one. Lowered to global_prefetch_b8 on gfx1250.
    if (j + 1 < O_PER_WAVE)
      __builtin_prefetch((const char*)(row + FFC_I) + lane * 128, 0, 0);

    const bool tnorm = (op_idx[o] == 0);  // wave-uniform at runtime
    float acc;

    if (tnorm) {
      acc = 1.0e30f;
      #pragma unroll
      for (int it = 0; it < FFC_I / (WAVE_SZ * 4); ++it) {
        const int idx = it * (WAVE_SZ * 4) + lane * 4;
        v4i e  = __builtin_nontemporal_load((const v4i*)(row + idx));
        v4f xv = *(const v4f*)(xs + idx);
        v4f ox = *(const v4f*)(oxs + idx);
        #pragma unroll
        for (int k = 0; k < 4; ++k) {
          float v = (e[k] == 2) ? ox[k] : xv[k];
          if (e[k] == 0) v += 10.0f;   // NO_EDGE_OFFSET_T_NORM
          acc = fminf(acc, v);
        }
      }
      acc = wave32_reduce_min(acc);
    } else {
      acc = -1.0e30f;
      #pragma unroll
      for (int it = 0; it < FFC_I / (WAVE_SZ * 4); ++it) {
        const int idx = it * (WAVE_SZ * 4) + lane * 4;
        v4i e  = __builtin_nontemporal_load((const v4i*)(row + idx));
        v4f xv = *(const v4f*)(xs + idx);
        v4f ox = *(const v4f*)(oxs + idx);
        #pragma unroll
        for (int k = 0; k < 4; ++k) {
          float v = (e[k] == 2) ? ox[k] : xv[k];
          if (e[k] == 0) v += -10.0f;  // NO_EDGE_OFFSET_T_CONORM
          acc = fmaxf(acc, v);
        }
      }
      acc = wave32_reduce_max(acc);
    }

    if (lane == 0) out[(size_t)b * FFC_O + o] = acc;
  }
}

extern "C" void kernel_launch(void* const* d_in, const int* in_sizes, int n_in,
                              void* d_out, int out_size, void* d_ws, size_t ws_size,
                              hipStream_t stream) {
  (void)in_sizes; (void)n_in; (void)out_size; (void)d_ws; (void)ws_size;
  const float* x      = (const float*)d_in[0];  // (B, I) f32
  const int*   et     = (const int*)d_in[1];    // (B, O, I) i32
  const int*   op_idx = (const int*)d_in[2];    // (O,) i32
  float*       out    = (float*)d_out;          // (B, O) f32

  dim3 grid(FFC_B, FFC_O / OBLK);   // (64, 64)
  dim3 block(BLOCK_THREADS);        // 8 wave32 waves
  ffc_minmax_kernel<<<grid, block, 0, stream>>>(x, et, op_idx, out);
}